// XConv_12713103197221
// MI455X (gfx1250) — compile-verified
//
#include <hip/hip_runtime.h>
#include <hip/hip_bf16.h>

// ---------------- problem constants ----------------
#define NN    65536
#define KK    16
#define DD    3
#define CIN   64
#define CDELT 16
#define COUT  128
#define CCAT  80           // 64 + 16
#define DMUL  2
#define EE    (NN * KK)    // 1,048,576 edges
#define EPSF  1e-5f

#define INV_E (1.0f / (float)EE)
#define INV_N (1.0f / (float)NN)

// ---------------- workspace layout (floats) ----------------
// stats region (zeroed every launch):
//   [0   ..16 )  mlp1 layer1 sum      [16 ..32 ) sumsq
//   [32  ..48 )  mlp1 layer2 sum      [48 ..64 ) sumsq
//   [64  ..320)  mlp2 A sum           [320..576) sumsq
//   [576 ..832)  mlp2 B sum           [832..1088) sumsq
//   [1088..1344) mlp2 C sum           [1344..1600) sumsq
#define ST_S1   0
#define ST_S2   32
#define ST_A    64
#define ST_B    576
#define ST_C    1088
#define WS_STATS 2048
#define WS_BUFA  WS_STATS                  // [N*256]
#define WS_BUFB  (WS_STATS + NN * 256)     // [N*256]; reused as y[N*160]

typedef float v2f __attribute__((ext_vector_type(2)));
typedef float v8f __attribute__((ext_vector_type(8)));

__device__ __forceinline__ float eluf(float v) {
    return v > 0.0f ? v : (__expf(v) - 1.0f);
}

// ---------------- kernel 0: zero the stats region ----------------
__global__ void k_zero_stats(float* ws) {
    int i = blockIdx.x * blockDim.x + threadIdx.x;
    if (i < WS_STATS) ws[i] = 0.0f;
}

// ---------------- kernel 1: mlp1 layer-1 stats ----------------
__global__ void k_mlp1_stats1(const int* __restrict__ ei,
                              const float* __restrict__ posFrom,
                              const float* __restrict__ posTo,
                              const float* __restrict__ W1,
                              const float* __restrict__ b1,
                              float* __restrict__ ws) {
    __shared__ float ls[32];
    if (threadIdx.x < 32) ls[threadIdx.x] = 0.0f;
    __syncthreads();
    int e = blockIdx.x * blockDim.x + threadIdx.x;
    if (e < EE) {
        int iF = ei[e], iT = ei[EE + e];
        float rp0 = posTo[iT * 3 + 0] - posFrom[iF * 3 + 0];
        float rp1 = posTo[iT * 3 + 1] - posFrom[iF * 3 + 1];
        float rp2 = posTo[iT * 3 + 2] - posFrom[iF * 3 + 2];
#pragma unroll
        for (int c = 0; c < 16; ++c) {
            float u = b1[c] + rp0 * W1[c] + rp1 * W1[16 + c] + rp2 * W1[32 + c];
            u = eluf(u);
            atomicAdd(&ls[c], u);
            atomicAdd(&ls[16 + c], u * u);
        }
    }
    __syncthreads();
    if (threadIdx.x < 32) atomicAdd(&ws[ST_S1 + threadIdx.x], ls[threadIdx.x]);
}

// ---------------- kernel 2: mlp1 layer-2 stats (recompute layer 1) -------
__global__ void k_mlp1_stats2(const int* __restrict__ ei,
                              const float* __restrict__ posFrom,
                              const float* __restrict__ posTo,
                              const float* __restrict__ W1, const float* __restrict__ b1,
                              const float* __restrict__ g1, const float* __restrict__ be1,
                              const float* __restrict__ W2, const float* __restrict__ b2,
                              float* __restrict__ ws) {
    __shared__ float ls[32];
    if (threadIdx.x < 32) ls[threadIdx.x] = 0.0f;
    __syncthreads();
    int e = blockIdx.x * blockDim.x + threadIdx.x;
    if (e < EE) {
        int iF = ei[e], iT = ei[EE + e];
        float rp0 = posTo[iT * 3 + 0] - posFrom[iF * 3 + 0];
        float rp1 = posTo[iT * 3 + 1] - posFrom[iF * 3 + 1];
        float rp2 = posTo[iT * 3 + 2] - posFrom[iF * 3 + 2];
        float h1[16];
#pragma unroll
        for (int c = 0; c < 16; ++c) {
            float u = b1[c] + rp0 * W1[c] + rp1 * W1[16 + c] + rp2 * W1[32 + c];
            u = eluf(u);
            float m = ws[ST_S1 + c] * INV_E;
            float v = ws[ST_S1 + 16 + c] * INV_E - m * m;
            h1[c] = (u - m) * rsqrtf(v + EPSF) * g1[c] + be1[c];
        }
#pragma unroll
        for (int j = 0; j < 16; ++j) {
            float u = b2[j];
#pragma unroll
            for (int c = 0; c < 16; ++c) u += h1[c] * W2[c * 16 + j];
            u = eluf(u);
            atomicAdd(&ls[j], u);
            atomicAdd(&ls[16 + j], u * u);
        }
    }
    __syncthreads();
    if (threadIdx.x < 32) atomicAdd(&ws[ST_S2 + threadIdx.x], ls[threadIdx.x]);
}

// ---------------- kernel 3: mlp2 linear 48->256 + elu + stats ----------
__global__ void k_mlp2_a(const int* __restrict__ ei,
                         const float* __restrict__ posFrom,
                         const float* __restrict__ posTo,
                         const float* __restrict__ Wl, const float* __restrict__ bl,
                         float* __restrict__ ws) {
    __shared__ float ls[512];
    for (int i = threadIdx.x; i < 512; i += blockDim.x) ls[i] = 0.0f;
    __syncthreads();
    int t = blockIdx.x * blockDim.x + threadIdx.x;
    int n = t >> 4, jg = t & 15;
    if (n < NN) {
        float rp[48];
#pragma unroll
        for (int k = 0; k < 16; ++k) {
            int e = n * 16 + k;
            int iF = ei[e], iT = ei[EE + e];
            rp[k * 3 + 0] = posTo[iT * 3 + 0] - posFrom[iF * 3 + 0];
            rp[k * 3 + 1] = posTo[iT * 3 + 1] - posFrom[iF * 3 + 1];
            rp[k * 3 + 2] = posTo[iT * 3 + 2] - posFrom[iF * 3 + 2];
        }
        float* bufA = ws + WS_BUFA;
#pragma unroll
        for (int jo = 0; jo < 16; ++jo) {
            int j = jg * 16 + jo;
            float u = bl[j];
#pragma unroll
            for (int r = 0; r < 48; ++r) u += rp[r] * Wl[r * 256 + j];
            u = eluf(u);
            bufA[n * 256 + j] = u;
            atomicAdd(&ls[j], u);
            atomicAdd(&ls[256 + j], u * u);
        }
    }
    __syncthreads();
    for (int i = threadIdx.x; i < 512; i += blockDim.x) atomicAdd(&ws[ST_A + i], ls[i]);
}

// ------- kernel 4: bnA -> grouped conv wA -> elu -> bufB, statsB --------
__global__ void k_mlp2_b(const float* __restrict__ wA, const float* __restrict__ biA,
                         const float* __restrict__ gA, const float* __restrict__ bA,
                         float* __restrict__ ws) {
    __shared__ float ls[512];
    for (int i = threadIdx.x; i < 512; i += blockDim.x) ls[i] = 0.0f;
    __syncthreads();
    int t = blockIdx.x * blockDim.x + threadIdx.x;
    int n = t >> 4, g = t & 15;
    if (n < NN) {
        const float* bufA = ws + WS_BUFA;
        float* bufB = ws + WS_BUFB;
        float tin[16];
#pragma unroll
        for (int tt = 0; tt < 16; ++tt) {
            int ch = g * 16 + tt;
            float m = ws[ST_A + ch] * INV_N;
            float v = ws[ST_A + 256 + ch] * INV_N - m * m;
            tin[tt] = (bufA[n * 256 + ch] - m) * rsqrtf(v + EPSF) * gA[ch] + bA[ch];
        }
#pragma unroll
        for (int j = 0; j < 16; ++j) {
            float u = biA[g * 16 + j];
#pragma unroll
            for (int tt = 0; tt < 16; ++tt) u += tin[tt] * wA[(g * 16 + j) * 16 + tt];
            u = eluf(u);
            bufB[n * 256 + g * 16 + j] = u;
            atomicAdd(&ls[g * 16 + j], u);
            atomicAdd(&ls[256 + g * 16 + j], u * u);
        }
    }
    __syncthreads();
    for (int i = threadIdx.x; i < 512; i += blockDim.x) atomicAdd(&ws[ST_B + i], ls[i]);
}

// ------- kernel 5: bnB -> grouped conv wB (no elu) -> bufA, statsC ------
__global__ void k_mlp2_c(const float* __restrict__ wB, const float* __restrict__ biB,
                         const float* __restrict__ gB, const float* __restrict__ bB,
                         float* __restrict__ ws) {
    __shared__ float ls[512];
    for (int i = threadIdx.x; i < 512; i += blockDim.x) ls[i] = 0.0f;
    __syncthreads();
    int t = blockIdx.x * blockDim.x + threadIdx.x;
    int n = t >> 4, g = t & 15;
    if (n < NN) {
        const float* bufB = ws + WS_BUFB;
        float* bufA = ws + WS_BUFA;
        float tin[16];
#pragma unroll
        for (int tt = 0; tt < 16; ++tt) {
            int ch = g * 16 + tt;
            float m = ws[ST_B + ch] * INV_N;
            float v = ws[ST_B + 256 + ch] * INV_N - m * m;
            tin[tt] = (bufB[n * 256 + ch] - m) * rsqrtf(v + EPSF) * gB[ch] + bB[ch];
        }
#pragma unroll
        for (int j = 0; j < 16; ++j) {
            float u = biB[g * 16 + j];
#pragma unroll
            for (int tt = 0; tt < 16; ++tt) u += tin[tt] * wB[(g * 16 + j) * 16 + tt];
            bufA[n * 256 + g * 16 + j] = u;
            atomicAdd(&ls[g * 16 + j], u);
            atomicAdd(&ls[256 + g * 16 + j], u * u);
        }
    }
    __syncthreads();
    for (int i = threadIdx.x; i < 512; i += blockDim.x) atomicAdd(&ws[ST_C + i], ls[i]);
}

// ------- kernel 6: per-point X-transform via f32 WMMA + depthwise -------
#define TWAVES 4
__global__ void __launch_bounds__(32 * TWAVES)
k_transform(const int* __restrict__ ei,
            const float* __restrict__ posFrom, const float* __restrict__ posTo,
            const float* __restrict__ x,
            const float* __restrict__ W1, const float* __restrict__ b1,
            const float* __restrict__ g1, const float* __restrict__ be1,
            const float* __restrict__ W2, const float* __restrict__ b2,
            const float* __restrict__ g2, const float* __restrict__ be2,
            const float* __restrict__ gC, const float* __restrict__ bC,
            const float* __restrict__ wD, const float* __restrict__ biD,
            const float* __restrict__ ws, float* __restrict__ y) {
    __shared__ float xs[TWAVES][16][84];  // x_star: rows j (16), cols c (80)
    __shared__ float xt[TWAVES][16][84];  // xt^T : rows k (16), cols c (80)
    const int lane = threadIdx.x & 31;
    const int w = threadIdx.x >> 5;
    const int n = blockIdx.x * TWAVES + w;
    const float* bufA = ws + WS_BUFA;

    // ---- A fragments: t[k][j] = bnC(bufA[n]), A layout M=lane%16, K=4s+2*(lane/16)+v
    v2f a[4];
    {
        const int row = lane & 15;
        const int koff = (lane >> 4) * 2;
#pragma unroll
        for (int s = 0; s < 4; ++s) {
#pragma unroll
            for (int v = 0; v < 2; ++v) {
                int col = s * 4 + koff + v;
                int ch = row * 16 + col;
                float m = ws[ST_C + ch] * INV_N;
                float va = ws[ST_C + 256 + ch] * INV_N - m * m;
                a[s][v] = (bufA[n * 256 + ch] - m) * rsqrtf(va + EPSF) * gC[ch] + bC[ch];
            }
        }
    }

    // ---- x_star rows: cols 0..15 = recomputed mlp1 h, cols 16..79 = gathered x
    if (lane < 16) {
        int e = n * 16 + lane;
        int iF = ei[e], iT = ei[EE + e];
        float rp0 = posTo[iT * 3 + 0] - posFrom[iF * 3 + 0];
        float rp1 = posTo[iT * 3 + 1] - posFrom[iF * 3 + 1];
        float rp2 = posTo[iT * 3 + 2] - posFrom[iF * 3 + 2];
        float h1[16];
#pragma unroll
        for (int c = 0; c < 16; ++c) {
            float u = b1[c] + rp0 * W1[c] + rp1 * W1[16 + c] + rp2 * W1[32 + c];
            u = eluf(u);
            float m = ws[ST_S1 + c] * INV_E;
            float v = ws[ST_S1 + 16 + c] * INV_E - m * m;
            h1[c] = (u - m) * rsqrtf(v + EPSF) * g1[c] + be1[c];
        }
#pragma unroll
        for (int j = 0; j < 16; ++j) {
            float u = b2[j];
#pragma unroll
            for (int c = 0; c < 16; ++c) u += h1[c] * W2[c * 16 + j];
            u = eluf(u);
            float m = ws[ST_S2 + j] * INV_E;
            float v = ws[ST_S2 + 16 + j] * INV_E - m * m;
            xs[w][lane][j] = (u - m) * rsqrtf(v + EPSF) * g2[j] + be2[j];
        }
    }
    {   // gather x rows: lane handles row lane/2, 32-col half (lane&1)
        int r = lane >> 1;
        int cbase = (lane & 1) * 32;
        int iF = ei[n * 16 + r];
        const float4* xp = (const float4*)(x + iF * 64 + cbase);
#pragma unroll
        for (int q = 0; q < 8; ++q) {
            float4 vv = xp[q];
            xs[w][r][16 + cbase + q * 4 + 0] = vv.x;
            xs[w][r][16 + cbase + q * 4 + 1] = vv.y;
            xs[w][r][16 + cbase + q * 4 + 2] = vv.z;
            xs[w][r][16 + cbase + q * 4 + 3] = vv.w;
        }
    }
    __syncthreads();

    // ---- xt^T[k][c] = sum_j t[k][j] * x_star[j][c] via V_WMMA_F32_16X16X4_F32
    const int bcol16 = lane & 15;
    const int koff = (lane >> 4) * 2;
    const int rbase = (lane >> 4) * 8;
#pragma unroll
    for (int ct = 0; ct < 5; ++ct) {
        v8f c = {};
#pragma unroll
        for (int s = 0; s < 4; ++s) {
            v2f b;
            int kr = s * 4 + koff;
            b[0] = xs[w][kr + 0][ct * 16 + bcol16];
            b[1] = xs[w][kr + 1][ct * 16 + bcol16];
            c = __builtin_amdgcn_wmma_f32_16x16x4_f32(false, a[s], false, b,
                                                      (short)0, c, false, false);
        }
#pragma unroll
        for (int v = 0; v < 8; ++v) xt[w][rbase + v][ct * 16 + bcol16] = c[v];
    }
    __syncthreads();

    // ---- depthwise conv: y[n, c*2+d] = sum_k xt^T[k][c]*wD[c][d][k] + biD
#pragma unroll
    for (int q = 0; q < 5; ++q) {
        int o = lane * 5 + q;        // 0..159
        int cc = o >> 1, d = o & 1;
        float acc = biD[cc * 2 + d];
#pragma unroll
        for (int k = 0; k < 16; ++k) acc += xt[w][k][cc] * wD[(cc * 2 + d) * 16 + k];
        y[n * 160 + o] = acc;
    }
}

// ------- kernel 7: out = y[N,160] @ Wf[160,128] + bf via f32 WMMA -------
__global__ void __launch_bounds__(256)
k_final(const float* __restrict__ y, const float* __restrict__ Wf,
        const float* __restrict__ bf, float* __restrict__ out) {
    __shared__ float At[16][164];    // 16 points x 160, padded
    const int lane = threadIdx.x & 31;
    const int w = threadIdx.x >> 5;  // col tile 0..7
    const int pt = blockIdx.x;       // point tile
    for (int i = threadIdx.x; i < 16 * 160; i += 256) {
        int r = i / 160, cidx = i - r * 160;
        At[r][cidx] = y[(pt * 16 + r) * 160 + cidx];
    }
    __syncthreads();

    const int arow = lane & 15;
    const int koff = (lane >> 4) * 2;
    const int bcol = w * 16 + (lane & 15);
    v8f c;
    float bias = bf[bcol];
#pragma unroll
    for (int v = 0; v < 8; ++v) c[v] = bias;
#pragma unroll 4
    for (int s = 0; s < 40; ++s) {
        v2f a, b;
        int kr = s * 4 + koff;
        a[0] = At[arow][kr + 0];
        a[1] = At[arow][kr + 1];
        b[0] = Wf[(kr + 0) * 128 + bcol];
        b[1] = Wf[(kr + 1) * 128 + bcol];
        c = __builtin_amdgcn_wmma_f32_16x16x4_f32(false, a, false, b,
                                                  (short)0, c, false, false);
    }
    const int rbase = (lane >> 4) * 8;
#pragma unroll
    for (int v = 0; v < 8; ++v)
        out[(pt * 16 + rbase + v) * 128 + bcol] = c[v];
}

// ---------------- launch ----------------
extern "C" void kernel_launch(void* const* d_in, const int* in_sizes, int n_in,
                              void* d_out, int out_size, void* d_ws, size_t ws_size,
                              hipStream_t stream) {
    (void)in_sizes; (void)n_in; (void)out_size; (void)ws_size;
    const float* x       = (const float*)d_in[0];
    const float* posFrom = (const float*)d_in[1];
    const float* posTo   = (const float*)d_in[2];
    const int*   ei      = (const int*)d_in[3];
    const float* W1 = (const float*)d_in[4];  const float* b1  = (const float*)d_in[5];
    const float* g1 = (const float*)d_in[6];  const float* be1 = (const float*)d_in[7];
    const float* W2 = (const float*)d_in[8];  const float* b2  = (const float*)d_in[9];
    const float* g2 = (const float*)d_in[10]; const float* be2 = (const float*)d_in[11];
    const float* Wl = (const float*)d_in[12]; const float* bl  = (const float*)d_in[13];
    const float* gA = (const float*)d_in[14]; const float* bA  = (const float*)d_in[15];
    const float* wA = (const float*)d_in[16]; const float* biA = (const float*)d_in[17];
    const float* gB = (const float*)d_in[18]; const float* bB  = (const float*)d_in[19];
    const float* wB = (const float*)d_in[20]; const float* biB = (const float*)d_in[21];
    const float* gC = (const float*)d_in[22]; const float* bC  = (const float*)d_in[23];
    const float* wD = (const float*)d_in[24]; const float* biD = (const float*)d_in[25];
    const float* Wf = (const float*)d_in[26]; const float* bf  = (const float*)d_in[27];
    float* ws  = (float*)d_ws;
    float* out = (float*)d_out;
    float* ybuf = ws + WS_BUFB;   // reuse bufB region for y[N,160]

    k_zero_stats<<<WS_STATS / 256, 256, 0, stream>>>(ws);
    k_mlp1_stats1<<<EE / 256, 256, 0, stream>>>(ei, posFrom, posTo, W1, b1, ws);
    k_mlp1_stats2<<<EE / 256, 256, 0, stream>>>(ei, posFrom, posTo, W1, b1, g1, be1,
                                                W2, b2, ws);
    k_mlp2_a<<<(NN * 16) / 256, 256, 0, stream>>>(ei, posFrom, posTo, Wl, bl, ws);
    k_mlp2_b<<<(NN * 16) / 256, 256, 0, stream>>>(wA, biA, gA, bA, ws);
    k_mlp2_c<<<(NN * 16) / 256, 256, 0, stream>>>(wB, biB, gB, bB, ws);
    k_transform<<<NN / TWAVES, 32 * TWAVES, 0, stream>>>(
        ei, posFrom, posTo, x, W1, b1, g1, be1, W2, b2, g2, be2,
        gC, bC, wD, biD, ws, ybuf);
    k_final<<<NN / 16, 256, 0, stream>>>(ybuf, Wf, bf, out);
}